// GLoss_21345987461643
// MI455X (gfx1250) — compile-verified
//
#include <hip/hip_runtime.h>
#include <hip/hip_bf16.h>

#define B_ 8
#define N_ 4096
#define M_ 512
#define EPS_ 0.01f
#define INV_EPS_ 100.0f
#define RHO_ 0.5f
#define DAMP_ (50.0f/51.0f)   // rho/(rho+eps)
#define TAU_ 0.1f
#define BLUR_ 0.01f
#define NEG_INF_ -3.0e38f

typedef __attribute__((ext_vector_type(2))) float v2f;
typedef __attribute__((ext_vector_type(8))) float v8f;

#if defined(__has_builtin)
#if __has_builtin(__builtin_amdgcn_wmma_f32_16x16x4_f32)
#define HAVE_WMMA_F32X4 1
#endif
#if __has_builtin(__builtin_amdgcn_sqrtf)
#define FAST_SQRT(x) __builtin_amdgcn_sqrtf(x)
#endif
#endif
#ifndef FAST_SQRT
#define FAST_SQRT(x) sqrtf(x)
#endif

// ---------------------------------------------------------------------------
// init: zero f, g, accumulators; precompute log_a = log(A + 1e-20)
// ---------------------------------------------------------------------------
__global__ __launch_bounds__(256) void init_kernel(const float* __restrict__ A,
                                                   float* __restrict__ f,
                                                   float* __restrict__ g,
                                                   float* __restrict__ la,
                                                   float* __restrict__ acc) {
    int i = blockIdx.x * 256 + threadIdx.x;        // grid covers B_*N_
    if (i < B_ * N_) {
        la[i] = __logf(A[i] + 1e-20f);
        f[i]  = 0.0f;
    }
    if (i < B_ * M_) g[i] = 0.0f;
    if (i < 8)       acc[i] = 0.0f;
}

// ---------------------------------------------------------------------------
// cost: Cs[b,n,m] = (exp(dist(X_n, Y_m)/0.6) - 1) / eps
// D2 tile built with V_WMMA_F32_16X16X4_F32:
//   A rows  = [xn, yn, |Xn|^2, 1]      (16x4)
//   B cols  = [-2xm, -2ym, 1, |Ym|^2]  (4x16)
//   D       = |Xn|^2 + |Ym|^2 - 2 Xn.Ym
// One wave per 16x16 tile; 8 waves / block.
// ---------------------------------------------------------------------------
__global__ __launch_bounds__(256) void cost_kernel(const float* __restrict__ pts,
                                                   float* __restrict__ Cs) {
    int wave = threadIdx.x >> 5;
    int lane = threadIdx.x & 31;
    int tile = blockIdx.x * 8 + wave;          // 8*256*32 = 65536 tiles
    int b  = tile >> 13;                       // / (256*32)
    int rm = tile & 8191;
    int nt = rm >> 5;                          // n-tile 0..255
    int mt = rm & 31;                          // m-tile 0..31
    int r  = lane & 15;
    int hi = lane >> 4;

    // grid coordinate: n = py*64+px, coord = (p+1)/64  (== ((p+0.5)*8+4)/512)
    int n = nt * 16 + r;
    float xn = (float)((n & 63) + 1) * (1.0f / 64.0f);
    float yn = (float)((n >> 6) + 1) * (1.0f / 64.0f);

    int m = mt * 16 + r;
    float xm = pts[((b << 9) + m) * 2 + 0] * (1.0f / 512.0f);
    float ym = pts[((b << 9) + m) * 2 + 1] * (1.0f / 512.0f);

    float d2v[8];
#ifdef HAVE_WMMA_F32X4
    v2f a, bb;
    a.x  = hi ? (xn * xn + yn * yn) : xn;      // K=2 : K=0
    a.y  = hi ? 1.0f                : yn;      // K=3 : K=1
    bb.x = hi ? 1.0f                : (-2.0f * xm);
    bb.y = hi ? (xm * xm + ym * ym) : (-2.0f * ym);
    v8f c = {0.f,0.f,0.f,0.f,0.f,0.f,0.f,0.f};
    c = __builtin_amdgcn_wmma_f32_16x16x4_f32(false, a, false, bb,
                                              (short)0, c, false, false);
    #pragma unroll
    for (int i = 0; i < 8; ++i) d2v[i] = c[i];
#else
    #pragma unroll
    for (int i = 0; i < 8; ++i) {
        int ni = nt * 16 + i + 8 * hi;
        float xi = (float)((ni & 63) + 1) * (1.0f / 64.0f);
        float yi = (float)((ni >> 6) + 1) * (1.0f / 64.0f);
        float dx = xi - xm, dy = yi - ym;
        d2v[i] = dx * dx + dy * dy;
    }
#endif
    size_t base = ((size_t)b * N_) * M_ + (size_t)(mt * 16 + r);
    #pragma unroll
    for (int i = 0; i < 8; ++i) {
        int ni = nt * 16 + i + 8 * hi;         // D VGPR i -> rows i, i+8
        float d2   = fmaxf(d2v[i], 0.0f) + 1e-12f;
        float dist = FAST_SQRT(d2);
        float val  = (__expf(dist * (1.0f / 0.6f)) - 1.0f) * INV_EPS_;
        Cs[base + (size_t)ni * M_] = val;
    }
}

// ---------------------------------------------------------------------------
// g-update: g[b,m] = -eps*damp * LSE_n( log_a + f/eps - C/eps )
// block = (b, 16 m's); tx = m lane (coalesced along M), ty = 16 n-strips.
// Online LSE with chunk-of-8 rescale (~1.1 exp/element).
// ---------------------------------------------------------------------------
__global__ __launch_bounds__(256) void g_update(const float* __restrict__ Cs,
                                                const float* __restrict__ la,
                                                const float* __restrict__ f,
                                                float* __restrict__ g) {
    __shared__ float h[N_];
    __shared__ float sm[16][16];
    __shared__ float ss[16][16];
    int b  = blockIdx.x >> 5;     // grid = 8*32
    int mt = blockIdx.x & 31;
    for (int i = threadIdx.x; i < N_; i += 256)
        h[i] = la[b * N_ + i] + f[b * N_ + i] * INV_EPS_;
    __syncthreads();

    int tx = threadIdx.x & 15;
    int ty = threadIdx.x >> 4;
    int m  = mt * 16 + tx;
    const float* cp = Cs + ((size_t)b * N_) * M_ + m;

    float mr = NEG_INF_, s = 0.0f;
    #pragma unroll 1
    for (int c0 = 0; c0 < 256; c0 += 8) {      // per thread: n = ty + 16*k
        if (c0 + 8 < 256) {                    // prefetch next chunk (L2 hit path)
            #pragma unroll
            for (int u = 0; u < 8; u += 2)
                __builtin_prefetch(cp + (size_t)(ty + 16 * (c0 + 8 + u)) * M_, 0, 1);
        }
        float t[8];
        float lm = NEG_INF_;
        #pragma unroll
        for (int u = 0; u < 8; ++u) {
            int nn = ty + 16 * (c0 + u);
            t[u] = h[nn] - cp[(size_t)nn * M_];
            lm = fmaxf(lm, t[u]);
        }
        float nm = fmaxf(mr, lm);
        float accv = 0.0f;
        #pragma unroll
        for (int u = 0; u < 8; ++u) accv += __expf(t[u] - nm);
        s  = s * __expf(mr - nm) + accv;
        mr = nm;
    }
    sm[ty][tx] = mr;
    ss[ty][tx] = s;
    __syncthreads();
    if (ty == 0) {
        float M0 = sm[0][tx], S0 = ss[0][tx];
        #pragma unroll
        for (int k = 1; k < 16; ++k) {
            float M1 = sm[k][tx], S1 = ss[k][tx];
            float nm = fmaxf(M0, M1);
            S0 = S0 * __expf(M0 - nm) + S1 * __expf(M1 - nm);
            M0 = nm;
        }
        g[b * M_ + m] = -EPS_ * DAMP_ * (M0 + __logf(S0));
    }
}

// ---------------------------------------------------------------------------
// f-update: f[b,n] = -eps*damp * LSE_m( g/eps - C/eps )
// block = (b, 64 n's); each wave32 reduces 8 rows, butterfly LSE merge.
// ---------------------------------------------------------------------------
__global__ __launch_bounds__(256) void f_update(const float* __restrict__ Cs,
                                                const float* __restrict__ g,
                                                float* __restrict__ f) {
    __shared__ float gi[M_];
    int b  = blockIdx.x >> 6;     // grid = 8*64
    int nt = blockIdx.x & 63;
    for (int i = threadIdx.x; i < M_; i += 256)
        gi[i] = g[b * M_ + i] * INV_EPS_;
    __syncthreads();

    int lane = threadIdx.x & 31;
    int wv   = threadIdx.x >> 5;
    #pragma unroll 1
    for (int j = 0; j < 8; ++j) {
        int n = nt * 64 + wv * 8 + j;
        const float* cp = Cs + ((size_t)b * N_ + n) * M_;
        if (j + 1 < 8)                          // prefetch next row's lines
            __builtin_prefetch(cp + M_ + lane * 4, 0, 1);
        float mr = NEG_INF_, s = 0.0f;
        #pragma unroll
        for (int c0 = 0; c0 < 16; c0 += 8) {   // m = lane + 32*k
            float t[8];
            float lm = NEG_INF_;
            #pragma unroll
            for (int u = 0; u < 8; ++u) {
                int mm = lane + 32 * (c0 + u);
                t[u] = gi[mm] - cp[mm];
                lm = fmaxf(lm, t[u]);
            }
            float nm = fmaxf(mr, lm);
            float accv = 0.0f;
            #pragma unroll
            for (int u = 0; u < 8; ++u) accv += __expf(t[u] - nm);
            s  = s * __expf(mr - nm) + accv;
            mr = nm;
        }
        #pragma unroll
        for (int off = 16; off > 0; off >>= 1) {
            float mo = __shfl_xor(mr, off, 32);
            float so = __shfl_xor(s, off, 32);
            float nm = fmaxf(mr, mo);
            s  = s * __expf(mr - nm) + so * __expf(mo - nm);
            mr = nm;
        }
        if (lane == 0)
            f[b * N_ + n] = -EPS_ * DAMP_ * (mr + __logf(s));
    }
}

// ---------------------------------------------------------------------------
// loss over rows: PI = exp(h[n] + g[m]/eps - C/eps), h = log_a + f/eps
//   pixel += |sum_m PI - A| ; entropy += (PI+1e-20)log(PI+1e-20)
//   emd   += rho*A*(1-exp(-f/rho))
// ---------------------------------------------------------------------------
__global__ __launch_bounds__(256) void loss_rows(const float* __restrict__ Cs,
                                                 const float* __restrict__ la,
                                                 const float* __restrict__ f,
                                                 const float* __restrict__ g,
                                                 const float* __restrict__ A,
                                                 float* __restrict__ acc) {
    __shared__ float gi[M_];
    int b  = blockIdx.x >> 6;
    int nt = blockIdx.x & 63;
    for (int i = threadIdx.x; i < M_; i += 256)
        gi[i] = g[b * M_ + i] * INV_EPS_;
    __syncthreads();

    int lane = threadIdx.x & 31;
    int wv   = threadIdx.x >> 5;
    float lpix = 0.0f, lent = 0.0f, lemd = 0.0f;
    #pragma unroll 1
    for (int j = 0; j < 8; ++j) {
        int n = nt * 64 + wv * 8 + j;
        float fv = f[b * N_ + n];
        float hn = la[b * N_ + n] + fv * INV_EPS_;
        const float* cp = Cs + ((size_t)b * N_ + n) * M_;
        float rs = 0.0f, es = 0.0f;
        for (int mm = lane; mm < M_; mm += 32) {
            float p = __expf(hn + gi[mm] - cp[mm]);
            rs += p;
            float q = p + 1e-20f;
            es += q * __logf(q);
        }
        #pragma unroll
        for (int off = 16; off > 0; off >>= 1) {
            rs += __shfl_xor(rs, off, 32);
            es += __shfl_xor(es, off, 32);
        }
        if (lane == 0) {
            float a = A[b * N_ + n];
            lpix += fabsf(rs - a);
            lent += es;
            lemd += RHO_ * a * (1.0f - __expf(-fv * (1.0f / RHO_)));
        }
    }
    if (lane == 0) {
        atomicAdd(&acc[0], lemd);
        atomicAdd(&acc[1], lpix);
        atomicAdd(&acc[3], lent);
    }
}

// ---------------------------------------------------------------------------
// loss over cols: point += |sum_n PI - 1| ; emd += rho*(1-exp(-g/rho))
// ---------------------------------------------------------------------------
__global__ __launch_bounds__(256) void loss_cols(const float* __restrict__ Cs,
                                                 const float* __restrict__ la,
                                                 const float* __restrict__ f,
                                                 const float* __restrict__ g,
                                                 float* __restrict__ acc) {
    __shared__ float h[N_];
    __shared__ float ps[16][16];
    int b  = blockIdx.x >> 5;
    int mt = blockIdx.x & 31;
    for (int i = threadIdx.x; i < N_; i += 256)
        h[i] = la[b * N_ + i] + f[b * N_ + i] * INV_EPS_;
    __syncthreads();

    int tx = threadIdx.x & 15;
    int ty = threadIdx.x >> 4;
    int m  = mt * 16 + tx;
    float gim = g[b * M_ + m] * INV_EPS_;
    const float* cp = Cs + ((size_t)b * N_) * M_ + m;

    float cs = 0.0f;
    #pragma unroll 1
    for (int k = 0; k < 256; ++k) {
        int nn = ty + 16 * k;
        cs += __expf(h[nn] + gim - cp[(size_t)nn * M_]);
    }
    ps[ty][tx] = cs;
    __syncthreads();
    if (ty == 0) {
        float tot = 0.0f;
        #pragma unroll
        for (int k = 0; k < 16; ++k) tot += ps[k][tx];
        float gv = g[b * M_ + m];
        atomicAdd(&acc[2], fabsf(tot - 1.0f));
        atomicAdd(&acc[0], RHO_ * (1.0f - __expf(-gv * (1.0f / RHO_))));
    }
}

__global__ void finalize_kernel(const float* __restrict__ acc,
                                float* __restrict__ out) {
    if (threadIdx.x == 0)
        out[0] = acc[0] + TAU_ * (acc[1] + acc[2])
               + BLUR_ * acc[3] * (1.0f / (float)(N_ * M_));
}

// ---------------------------------------------------------------------------
extern "C" void kernel_launch(void* const* d_in, const int* in_sizes, int n_in,
                              void* d_out, int out_size, void* d_ws, size_t ws_size,
                              hipStream_t stream) {
    const float* A   = (const float*)d_in[0];   // predict_map [8,1,64,64]
    const float* pts = (const float*)d_in[1];   // points [8,512,2]

    const size_t nCs = (size_t)B_ * N_ * M_;    // 16,777,216 floats
    const size_t need = (nCs + B_ * N_ + B_ * M_ + B_ * N_ + 8) * sizeof(float);
    if (ws_size < need) return;                 // workspace too small

    float* Cs  = (float*)d_ws;
    float* f   = Cs + nCs;
    float* g   = f + (size_t)B_ * N_;
    float* la  = g + (size_t)B_ * M_;
    float* acc = la + (size_t)B_ * N_;

    init_kernel<<<(B_ * N_) / 256, 256, 0, stream>>>(A, f, g, la, acc);
    cost_kernel<<<8192, 256, 0, stream>>>(pts, Cs);
    for (int it = 0; it < 30; ++it) {
        g_update<<<B_ * (M_ / 16), 256, 0, stream>>>(Cs, la, f, g);
        f_update<<<B_ * (N_ / 64), 256, 0, stream>>>(Cs, g, f);
    }
    loss_rows<<<B_ * (N_ / 64), 256, 0, stream>>>(Cs, la, f, g, A, acc);
    loss_cols<<<B_ * (M_ / 16), 256, 0, stream>>>(Cs, la, f, g, acc);
    finalize_kernel<<<1, 32, 0, stream>>>(acc, (float*)d_out);
}